// Bond2BondLayer_5119601016921
// MI455X (gfx1250) — compile-verified
//
#include <hip/hip_runtime.h>
#include <cstddef>
#include <cstdint>

typedef float v2f __attribute__((ext_vector_type(2)));
typedef float v8f __attribute__((ext_vector_type(8)));

#define HID 128
#define TILE_M 64
#define LPAD 4            // LDS row stride HID+4 = 132 dwords (== 4 mod 64 banks): conflict-free frags

__device__ __forceinline__ float lrelu(float x) { return x > 0.f ? x : 0.01f * x; }

// ---------------- zero fill ----------------
__global__ void fill_zero_kernel(float* __restrict__ p, size_t n) {
    size_t i = (size_t)blockIdx.x * blockDim.x + threadIdx.x;
    size_t stride = (size_t)gridDim.x * blockDim.x;
    for (; i < n; i += stride) p[i] = 0.0f;
}

// ---------------- angle MLP table: 6 classes x 128 (collapses T-sized MLP) ----------------
// order in reference: relu(emb @ ang_in) -> relu(@ ang2) -> relu(@ ang1)
__global__ __launch_bounds__(128)
void ang_table_kernel(const float* __restrict__ emb,
                      const float* __restrict__ w_in, const float* __restrict__ b_in,
                      const float* __restrict__ w2, const float* __restrict__ b2,
                      const float* __restrict__ w1, const float* __restrict__ b1,
                      float* __restrict__ table) {
    __shared__ float cur[HID];
    const int j = threadIdx.x;
    for (int c = 0; c < 6; ++c) {
        cur[j] = emb[c * HID + j];
        __syncthreads();
        float s = b_in[j];
        for (int k = 0; k < HID; ++k) s += cur[k] * w_in[k * HID + j];
        s = fmaxf(s, 0.f);
        __syncthreads(); cur[j] = s; __syncthreads();
        s = b2[j];
        for (int k = 0; k < HID; ++k) s += cur[k] * w2[k * HID + j];
        s = fmaxf(s, 0.f);
        __syncthreads(); cur[j] = s; __syncthreads();
        s = b1[j];
        for (int k = 0; k < HID; ++k) s += cur[k] * w1[k * HID + j];
        s = fmaxf(s, 0.f);
        table[c * HID + j] = s;
        __syncthreads();
    }
}

// ---------------- fp32 WMMA GEMM: out[M,128] = epi(A[M,128] @ W[128,128] + b) ----------------
// W staged transposed in LDS so B fragments are contiguous b64 loads.
// epi: optional relu, then optional "+ res[row,col]" (matching he + relu(...)).
template <int ACT, int RES>
__global__ __launch_bounds__(256)
void gemm128_wmma_kernel(const float* __restrict__ A, const float* __restrict__ W,
                         const float* __restrict__ bias, const float* __restrict__ res,
                         float* __restrict__ out, int M) {
    __shared__ __align__(16) float sA[TILE_M][HID + LPAD];   // ~33 KB
    __shared__ __align__(16) float sWt[HID][HID + LPAD];     // ~66 KB, [n][k] (transposed)
    const int tid = threadIdx.x;
    const int m0 = blockIdx.x * TILE_M;
    {
        const float4* Wv = (const float4*)W;
        #pragma unroll
        for (int i = 0; i < (HID * HID / 4) / 256; ++i) {
            const int idx = tid + i * 256;
            const int k = idx >> 5;              // W row (K)
            const int n = (idx & 31) * 4;        // W col (N)
            const float4 w = Wv[idx];
            sWt[n + 0][k] = w.x;
            sWt[n + 1][k] = w.y;
            sWt[n + 2][k] = w.z;
            sWt[n + 3][k] = w.w;
        }
    }
    {
        const float4* Av = (const float4*)A;
        #pragma unroll
        for (int i = 0; i < (TILE_M * HID / 4) / 256; ++i) {
            const int idx = tid + i * 256;
            const int row = m0 + (idx >> 5);     // 32 float4 per row
            float4 v = make_float4(0.f, 0.f, 0.f, 0.f);
            if (row < M) v = Av[(size_t)row * (HID / 4) + (idx & 31)];
            *(float4*)&sA[idx >> 5][(idx & 31) * 4] = v;
        }
    }
    __syncthreads();

    const int wave = tid >> 5;
    const int lane = tid & 31;
    const int l16 = lane & 15;
    const int hi = lane >> 4;                 // 0: K0/K1 (rows M=r), 1: K2/K3 (rows M=r+8)
    const int rsel = (wave & 3) * 16 + l16;   // A row within tile held by this lane
    const int cbase = (wave >> 2) * 64;       // N base for this wave

    v8f acc[4] = {};
    #pragma unroll 4
    for (int kk = 0; kk < HID; kk += 4) {
        const v2f a = *(const v2f*)&sA[rsel][kk + 2 * hi];   // A frag: vgpr0=K0|K2, vgpr1=K1|K3
        #pragma unroll
        for (int n = 0; n < 4; ++n) {
            const v2f b = *(const v2f*)&sWt[cbase + n * 16 + l16][kk + 2 * hi];
            acc[n] = __builtin_amdgcn_wmma_f32_16x16x4_f32(
                false, a, false, b, (short)0, acc[n], false, false);
        }
    }

    #pragma unroll
    for (int n = 0; n < 4; ++n) {
        const int col = cbase + n * 16 + l16;
        const float bv = bias[col];
        #pragma unroll
        for (int r = 0; r < 8; ++r) {
            const int row = m0 + (wave & 3) * 16 + r + 8 * hi;   // C/D: vgpr r -> M=r / r+8
            if (row < M) {
                float v = acc[n][r] + bv;
                if (ACT) v = fmaxf(v, 0.f);
                if (RES) v += res[(size_t)row * HID + col];
                out[(size_t)row * HID + col] = v;
            }
        }
    }
}

static inline void launch_gemm(const float* A, const float* W, const float* b,
                               const float* res, float* out, int M, int act,
                               int grid, hipStream_t s) {
    if (act) {
        if (res) gemm128_wmma_kernel<1, 1><<<grid, 256, 0, s>>>(A, W, b, res, out, M);
        else     gemm128_wmma_kernel<1, 0><<<grid, 256, 0, s>>>(A, W, b, nullptr, out, M);
    } else {
        if (res) gemm128_wmma_kernel<0, 1><<<grid, 256, 0, s>>>(A, W, b, res, out, M);
        else     gemm128_wmma_kernel<0, 0><<<grid, 256, 0, s>>>(A, W, b, nullptr, out, M);
    }
}

// ---------------- attention pass A: per-triplet angle + logits + segment denominator ----------------
__global__ __launch_bounds__(256)
void attn_pass_a_kernel(const float* __restrict__ pos, const float* __restrict__ ddv,
                        const float* __restrict__ Kf, const float* __restrict__ Qf,
                        const float* __restrict__ aeT, const float* __restrict__ attnW,
                        const int* __restrict__ ikj, const int* __restrict__ iji,
                        const int* __restrict__ ii, const int* __restrict__ ij,
                        const int* __restrict__ ik,
                        float* __restrict__ att_all, float* __restrict__ attdec, int T) {
    const int t = (int)(((size_t)blockIdx.x * blockDim.x + threadIdx.x) >> 5);
    const int lane = threadIdx.x & 31;
    if (t >= T) return;
    const int ekj = ikj[t], eji = iji[t];
    const int ai = ii[t] * 3, aj = ij[t] * 3, ak = ik[t] * 3;
    const float pix = pos[ai], piy = pos[ai + 1], piz = pos[ai + 2];
    const float jx = pos[aj] - pix, jy = pos[aj + 1] - piy, jz = pos[aj + 2] - piz;
    const float kx = pos[ak] - pix, ky = pos[ak + 1] - piy, kz = pos[ak + 2] - piz;
    const float adot = jx * kx + jy * ky + jz * kz;
    const float cx = jy * kz - jz * ky;
    const float cy = jz * kx - jx * kz;
    const float cz = jx * ky - jy * kx;
    const float bn = sqrtf(cx * cx + cy * cy + cz * cz);
    const float angle = atan2f(bn, adot);
    int cls = (int)(angle * (6.0f / 3.1415926f));
    cls = cls < 0 ? 0 : (cls > 5 ? 5 : cls);   // jax gather clamps OOB (angle==pi case)

    const int f0 = lane * 4;                   // lane owns 4 consecutive feats; head = lane>>2
    const float4 kj = *(const float4*)(Kf + (size_t)ekj * HID + f0);
    const float4 qj = *(const float4*)(Qf + (size_t)eji * HID + f0);
    const float4 ae = *(const float4*)(aeT + (size_t)cls * HID + f0);
    const float4 aw = *(const float4*)(attnW + f0);
    float s = lrelu(kj.x + qj.x + ae.x) * aw.x
            + lrelu(kj.y + qj.y + ae.y) * aw.y
            + lrelu(kj.z + qj.z + ae.z) * aw.z
            + lrelu(kj.w + qj.w + ae.w) * aw.w;
    s += __shfl_xor(s, 1, 32);                 // reduce the 4 lanes of each head
    s += __shfl_xor(s, 2, 32);
    const float w = expf(s + ddv[ekj]);
    if ((lane & 3) == 0) {
        const int h = lane >> 2;
        attdec[(size_t)t * 8 + h] = w;
        atomicAdd(att_all + (size_t)eji * 8 + h, w);
    }
}

// ---------------- attention pass B: normalized weighted segment-sum into v[E,128] ----------------
__global__ __launch_bounds__(256)
void attn_pass_b_kernel(const float* __restrict__ Kf, const float* __restrict__ attdec,
                        const float* __restrict__ att_all,
                        const int* __restrict__ ikj, const int* __restrict__ iji,
                        float* __restrict__ v, int T) {
    const int t = (int)(((size_t)blockIdx.x * blockDim.x + threadIdx.x) >> 5);
    const int lane = threadIdx.x & 31;
    if (t >= T) return;
    const int ekj = ikj[t], eji = iji[t];
    const int h = lane >> 2;
    const float norm = attdec[(size_t)t * 8 + h] / att_all[(size_t)eji * 8 + h];
    const int f0 = lane * 4;
    const float4 kj = *(const float4*)(Kf + (size_t)ekj * HID + f0);
    float* dst = v + (size_t)eji * HID + f0;
    atomicAdd(dst + 0, kj.x * norm);
    atomicAdd(dst + 1, kj.y * norm);
    atomicAdd(dst + 2, kj.z * norm);
    atomicAdd(dst + 3, kj.w * norm);
}

extern "C" void kernel_launch(void* const* d_in, const int* in_sizes, int n_in,
                              void* d_out, int out_size, void* d_ws, size_t ws_size,
                              hipStream_t stream) {
    (void)n_in; (void)out_size; (void)ws_size;
    const float* pos  = (const float*)d_in[0];
    const float* ddv  = (const float*)d_in[1];
    const float* bond = (const float*)d_in[2];
    const int* ikj = (const int*)d_in[3];
    const int* iji = (const int*)d_in[4];
    const int* ii  = (const int*)d_in[5];
    const int* ij  = (const int*)d_in[6];
    const int* ik  = (const int*)d_in[7];
    // params flattened in make_params insertion order (24 leaves -> d_in[8..31])
    const float* k_w    = (const float*)d_in[8];
    const float* k_b    = (const float*)d_in[9];
    const float* q_w    = (const float*)d_in[10];
    const float* q_b    = (const float*)d_in[11];
    const float* attnW  = (const float*)d_in[12];
    const float* lin1_w = (const float*)d_in[13];
    const float* lin1_b = (const float*)d_in[14];
    const float* lin2_w = (const float*)d_in[15];
    const float* lin2_b = (const float*)d_in[16];
    const float* r1a_w  = (const float*)d_in[17];
    const float* r1a_b  = (const float*)d_in[18];
    const float* r1b_w  = (const float*)d_in[19];
    const float* r1b_b  = (const float*)d_in[20];
    const float* r2a_w  = (const float*)d_in[21];
    const float* r2a_b  = (const float*)d_in[22];
    const float* r2b_w  = (const float*)d_in[23];
    const float* r2b_b  = (const float*)d_in[24];
    const float* ang_emb  = (const float*)d_in[25];
    const float* ang_in_w = (const float*)d_in[26];
    const float* ang_in_b = (const float*)d_in[27];
    const float* ang1_w   = (const float*)d_in[28];
    const float* ang1_b   = (const float*)d_in[29];
    const float* ang2_w   = (const float*)d_in[30];
    const float* ang2_b   = (const float*)d_in[31];

    const int E = in_sizes[1];   // dist_decay
    const int T = in_sizes[3];   // index_kj
    float* out = (float*)d_out;

    // workspace: bufK, bufQ (reused as V then as he/ping-pong), att_all[E,8], attdec[T,8], aeT[6,128]
    float* bufK    = (float*)d_ws;
    float* bufQ    = bufK + (size_t)E * HID;
    float* att_all = bufQ + (size_t)E * HID;
    float* attdec  = att_all + (size_t)E * 8;
    float* aeT     = attdec + (size_t)T * 8;

    const int gGemm = (E + TILE_M - 1) / TILE_M;
    const int gTrip = (T + 7) / 8;   // one wave32 per triplet, 8 waves per block

    // 1. collapse the angle MLP to a 6x128 table
    ang_table_kernel<<<1, 128, 0, stream>>>(ang_emb, ang_in_w, ang_in_b,
                                            ang2_w, ang2_b, ang1_w, ang1_b, aeT);
    // 2. K/Q projections of edges
    launch_gemm(bond, k_w, k_b, nullptr, bufK, E, 0, gGemm, stream);
    launch_gemm(bond, q_w, q_b, nullptr, bufQ, E, 0, gGemm, stream);
    // 3. segment-softmax denominators
    fill_zero_kernel<<<1024, 256, 0, stream>>>(att_all, (size_t)E * 8);
    attn_pass_a_kernel<<<gTrip, 256, 0, stream>>>(pos, ddv, bufK, bufQ, aeT, attnW,
                                                  ikj, iji, ii, ij, ik, att_all, attdec, T);
    // 4. weighted segment-sum v (bufQ no longer needed -> reuse as V)
    fill_zero_kernel<<<2048, 256, 0, stream>>>(bufQ, (size_t)E * HID);
    attn_pass_b_kernel<<<gTrip, 256, 0, stream>>>(bufK, attdec, att_all, ikj, iji, bufQ, T);
    // 5. he = relu(v@lin1+b)@lin2 + b + bond
    launch_gemm(bufQ, lin1_w, lin1_b, nullptr, bufK, E, 1, gGemm, stream);
    launch_gemm(bufK, lin2_w, lin2_b, bond, bufQ, E, 0, gGemm, stream);
    // 6. residual block 1: he += relu(relu(he@r1a+b)@r1b+b)
    launch_gemm(bufQ, r1a_w, r1a_b, nullptr, bufK, E, 1, gGemm, stream);
    launch_gemm(bufK, r1b_w, r1b_b, bufQ, bufQ, E, 1, gGemm, stream);
    // 7. residual block 2 -> d_out
    launch_gemm(bufQ, r2a_w, r2a_b, nullptr, bufK, E, 1, gGemm, stream);
    launch_gemm(bufK, r2b_w, r2b_b, bufQ, out, E, 1, gGemm, stream);
}